// NetGCN_41377714930256
// MI455X (gfx1250) — compile-verified
//
#include <hip/hip_runtime.h>
#include <hip/hip_bf16.h>

typedef __attribute__((ext_vector_type(2))) float v2f;
typedef __attribute__((ext_vector_type(8))) float v8f;

#define NN 50000
#define EE 600000
#define DIN 128
#define DE 64
#define HH 128

// ---------------------------------------------------------------------------
// Generic WMMA f32 GEMM:  C[M,128] = op(A)[M,K] @ W[K,128] + bias (+ addsrc)
//   op(A): optional pair-gather (A row m = Abase[idx_a[m]] + Abase[idx_b[m]])
//          then optional fused BN-ReLU normalization:
//          v = relu(v) * scale[k] + shift[k]
// Block: 256 threads = 8 waves; block computes 16 rows x 128 cols.
// A tile staged in LDS (padded), B fragments streamed from global (L2-hot).
// Uses v_wmma_f32_16x16x4_f32 (full fp32 accuracy; compute is not the
// bottleneck here so no reason to drop precision).
// ---------------------------------------------------------------------------
__global__ __launch_bounds__(256)
void gemm_wmma_f32(const float* __restrict__ A, const float* __restrict__ W,
                   const float* __restrict__ bias,
                   const float* __restrict__ scale, const float* __restrict__ shift,
                   const int* __restrict__ idx_a, const int* __restrict__ idx_b,
                   const float* __restrict__ addsrc,
                   float* __restrict__ C, int M, int K) {
  __shared__ float lds_a[16 * (128 + 4)];  // padded to kill bank conflicts
  const int lda = K + 4;
  const int t = threadIdx.x;
  const int m0 = blockIdx.x * 16;
  if (m0 >= M) return;

  // ---- stage A tile (16 x K) into LDS, fusing gather-pair + BN-ReLU ----
  for (int i = t; i < 16 * K; i += 256) {
    int r = i / K, k = i - r * K;
    int m = m0 + r;
    float v;
    if (idx_a) {
      v = A[(size_t)idx_a[m] * K + k] + A[(size_t)idx_b[m] * K + k];
    } else {
      v = A[(size_t)m * K + k];
    }
    if (scale) v = fmaxf(v, 0.0f) * scale[k] + shift[k];
    lds_a[r * lda + k] = v;
  }
  __syncthreads();

  const int lane = t & 31;
  const int wave = t >> 5;       // 0..7 -> 16-column block
  const int n0 = wave * 16;
  const int lr = lane & 15;      // row-in-half / col-in-half
  const int hi = lane >> 4;      // 0: K{0,1}, 1: K{2,3} half

  v8f acc = {};
  for (int k = 0; k < K; k += 4) {
    int ka = k + hi * 2;
    v2f a, b;
    a.x = lds_a[lr * lda + ka];
    a.y = lds_a[lr * lda + ka + 1];
    b.x = W[(size_t)ka * HH + n0 + lr];
    b.y = W[(size_t)(ka + 1) * HH + n0 + lr];
    acc = __builtin_amdgcn_wmma_f32_16x16x4_f32(
        /*neg_a=*/false, a, /*neg_b=*/false, b,
        /*c_mod=*/(short)0, acc, /*reuse_a=*/false, /*reuse_b=*/false);
  }

  // ---- epilogue: C layout VGPR j -> row (hi*8+j), col lr ----
  const int n = n0 + lr;
  const float bn_ = bias[n];
#pragma unroll
  for (int j = 0; j < 8; ++j) {
    int m = m0 + hi * 8 + j;
    float v = acc[j] + bn_;
    if (addsrc) v += addsrc[(size_t)m * HH + n];
    C[(size_t)m * HH + n] = v;
  }
}

// ---------------------------------------------------------------------------
// Fused message + scatter:  out[col[e]] += dis[row[e]]*dis[col[e]] *
//                                          relu(xh[row[e]] + eh[e])
// 8 edges per 256-thread block; one wave per edge, float4 per lane (128 ch).
// ---------------------------------------------------------------------------
__global__ __launch_bounds__(256)
void scatter_msg(const float* __restrict__ xh, const float* __restrict__ eh,
                 const int* __restrict__ row, const int* __restrict__ col,
                 const float* __restrict__ dis, float* __restrict__ out, int E_) {
  int e = blockIdx.x * 8 + (threadIdx.x >> 5);
  if (e >= E_) return;
  int lane = threadIdx.x & 31;
  int r = row[e], c = col[e];
  float nrm = dis[r] * dis[c];
  const float4* xv = (const float4*)(xh + (size_t)r * HH);
  const float4* ev = (const float4*)(eh + (size_t)e * HH);
  float4 a = xv[lane];
  float4 b = ev[lane];
  float* o = out + (size_t)c * HH + lane * 4;
  atomicAdd(o + 0, fmaxf(a.x + b.x, 0.0f) * nrm);
  atomicAdd(o + 1, fmaxf(a.y + b.y, 0.0f) * nrm);
  atomicAdd(o + 2, fmaxf(a.z + b.z, 0.0f) * nrm);
  atomicAdd(o + 3, fmaxf(a.w + b.w, 0.0f) * nrm);
}

// ---------------------------------------------------------------------------
// Degree: count occurrences of row indices (+1 later), dis = rsqrt(deg+1)
// ---------------------------------------------------------------------------
__global__ void deg_count(const int* __restrict__ row, float* __restrict__ deg, int E_) {
  int e = blockIdx.x * blockDim.x + threadIdx.x;
  if (e < E_) atomicAdd(&deg[row[e]], 1.0f);
}

__global__ void deg_finalize(float* __restrict__ dis, int n) {
  int i = blockIdx.x * blockDim.x + threadIdx.x;
  if (i < n) dis[i] = rsqrtf(dis[i] + 1.0f);
}

// ---------------------------------------------------------------------------
// BN stats over relu(X): per-channel sum and sum-of-squares (128 channels).
// 256 threads = 2 row-phases x 128 channels; partial -> LDS -> global atomics.
// ---------------------------------------------------------------------------
__global__ __launch_bounds__(256)
void bn_stats(const float* __restrict__ X, float* __restrict__ sums,
              int M, int rowsPerBlock) {
  __shared__ float s_sum[256];
  __shared__ float s_sq[256];
  int t = threadIdx.x;
  int c = t & 127;
  int h = t >> 7;
  size_t rend = (size_t)(blockIdx.x + 1) * rowsPerBlock;
  if (rend > (size_t)M) rend = (size_t)M;
  float s = 0.0f, q = 0.0f;
  for (size_t r = (size_t)blockIdx.x * rowsPerBlock + h; r < rend; r += 2) {
    float v = fmaxf(X[r * HH + c], 0.0f);
    s += v;
    q += v * v;
  }
  s_sum[t] = s;
  s_sq[t] = q;
  __syncthreads();
  if (h == 0) {
    atomicAdd(&sums[c], s_sum[c] + s_sum[c + 128]);
    atomicAdd(&sums[128 + c], s_sq[c] + s_sq[c + 128]);
  }
}

// scale = gamma * rsqrt(var+eps) ; shift = beta - mu*scale
__global__ void bn_finalize(const float* __restrict__ sums,
                            const float* __restrict__ gamma,
                            const float* __restrict__ beta,
                            float* __restrict__ scale, float* __restrict__ shift,
                            float Mf) {
  int c = threadIdx.x;
  float mu = sums[c] / Mf;
  float var = sums[128 + c] / Mf - mu * mu;
  float inv = rsqrtf(var + 1e-5f);
  float sc = gamma[c] * inv;
  scale[c] = sc;
  shift[c] = beta[c] - mu * sc;
}

// ---------------------------------------------------------------------------
// Host-side orchestration
// ---------------------------------------------------------------------------
extern "C" void kernel_launch(void* const* d_in, const int* in_sizes, int n_in,
                              void* d_out, int out_size, void* d_ws, size_t ws_size,
                              hipStream_t stream) {
  // ---- inputs (setup_inputs dict order, params flattened depth-first) ----
  const float* x         = (const float*)d_in[0];   // [N, DIN]
  const float* edge_attr = (const float*)d_in[1];   // [E, DE]
  const int*   eidx      = (const int*)d_in[2];     // [2, E]
  const float* W1   = (const float*)d_in[3];
  const float* b1   = (const float*)d_in[4];
  const float* We1  = (const float*)d_in[5];
  const float* be1  = (const float*)d_in[6];
  const float* W2   = (const float*)d_in[7];
  const float* b2   = (const float*)d_in[8];
  const float* We2  = (const float*)d_in[9];
  const float* be2  = (const float*)d_in[10];
  const float* W3   = (const float*)d_in[11];
  const float* b3   = (const float*)d_in[12];
  const float* We3  = (const float*)d_in[13];
  const float* be3  = (const float*)d_in[14];
  const float* Wu   = (const float*)d_in[15];
  const float* bu   = (const float*)d_in[16];
  const float* g1   = (const float*)d_in[17];
  const float* bt1  = (const float*)d_in[18];
  const float* g2   = (const float*)d_in[19];
  const float* bt2  = (const float*)d_in[20];
  const float* g3   = (const float*)d_in[21];
  const float* bt3  = (const float*)d_in[22];
  const float* ge   = (const float*)d_in[23];
  const float* bte  = (const float*)d_in[24];
  const float* Wn   = (const float*)d_in[25];
  const float* bn_b = (const float*)d_in[26];
  const float* Weo  = (const float*)d_in[27];
  const float* beo  = (const float*)d_in[28];

  const int* rowi = eidx;        // edge_index[0]
  const int* coli = eidx + EE;   // edge_index[1]

  float* x_out = (float*)d_out;            // [N, 128]
  float* e_out = x_out + (size_t)NN * HH;  // [E, 128]

  // ---- workspace layout ----
  float* bufA  = (float*)d_ws;                 // [N,128] xh scratch
  float* bufB  = bufA + (size_t)NN * HH;       // [N,128] conv output / x_raw
  float* eh    = bufB + (size_t)NN * HH;       // [E,128] eh (conv1/2), new_e
  float* e3    = eh   + (size_t)EE * HH;       // [E,128] conv3 edge embedding
  float* dis   = e3   + (size_t)EE * HH;       // [N]
  float* stats = dis  + NN;                    // [256]
  float* scale = stats + 256;                  // [128]
  float* shift = scale + 128;                  // [128]

  const int gemmThreads = 256;
  const dim3 gN(NN / 16), gE(EE / 16);
  const dim3 gScat(EE / 8);

  // ---- degrees ----
  hipMemsetAsync(dis, 0, NN * sizeof(float), stream);
  deg_count<<<(EE + 255) / 256, 256, 0, stream>>>(rowi, dis, EE);
  deg_finalize<<<(NN + 255) / 256, 256, 0, stream>>>(dis, NN);

  // ================= conv1 =================
  gemm_wmma_f32<<<gN, gemmThreads, 0, stream>>>(x, W1, b1, nullptr, nullptr,
                                                nullptr, nullptr, nullptr,
                                                bufA, NN, DIN);
  gemm_wmma_f32<<<gE, gemmThreads, 0, stream>>>(edge_attr, We1, be1, nullptr, nullptr,
                                                nullptr, nullptr, nullptr,
                                                eh, EE, DE);
  hipMemsetAsync(bufB, 0, (size_t)NN * HH * sizeof(float), stream);
  scatter_msg<<<gScat, 256, 0, stream>>>(bufA, eh, rowi, coli, dis, bufB, EE);
  // bn1 stats on relu(x1_raw)
  hipMemsetAsync(stats, 0, 256 * sizeof(float), stream);
  bn_stats<<<(NN + 127) / 128, 256, 0, stream>>>(bufB, stats, NN, 128);
  bn_finalize<<<1, 128, 0, stream>>>(stats, g1, bt1, scale, shift, (float)NN);

  // ================= conv2 =================  (bn1 fused into A-stage)
  gemm_wmma_f32<<<gN, gemmThreads, 0, stream>>>(bufB, W2, b2, scale, shift,
                                                nullptr, nullptr, nullptr,
                                                bufA, NN, HH);
  gemm_wmma_f32<<<gE, gemmThreads, 0, stream>>>(edge_attr, We2, be2, nullptr, nullptr,
                                                nullptr, nullptr, nullptr,
                                                eh, EE, DE);
  hipMemsetAsync(bufB, 0, (size_t)NN * HH * sizeof(float), stream);
  scatter_msg<<<gScat, 256, 0, stream>>>(bufA, eh, rowi, coli, dis, bufB, EE);
  hipMemsetAsync(stats, 0, 256 * sizeof(float), stream);
  bn_stats<<<(NN + 127) / 128, 256, 0, stream>>>(bufB, stats, NN, 128);
  bn_finalize<<<1, 128, 0, stream>>>(stats, g2, bt2, scale, shift, (float)NN);

  // ================= conv3 =================  (bn2 fused into A-stage)
  gemm_wmma_f32<<<gN, gemmThreads, 0, stream>>>(bufB, W3, b3, scale, shift,
                                                nullptr, nullptr, nullptr,
                                                bufA, NN, HH);
  gemm_wmma_f32<<<gE, gemmThreads, 0, stream>>>(edge_attr, We3, be3, nullptr, nullptr,
                                                nullptr, nullptr, nullptr,
                                                e3, EE, DE);
  hipMemsetAsync(bufB, 0, (size_t)NN * HH * sizeof(float), stream);
  scatter_msg<<<gScat, 256, 0, stream>>>(bufA, e3, rowi, coli, dis, bufB, EE);
  // bufB = x3_raw

  // ---- bn3 stats on relu(x3_raw) ----
  hipMemsetAsync(stats, 0, 256 * sizeof(float), stream);
  bn_stats<<<(NN + 127) / 128, 256, 0, stream>>>(bufB, stats, NN, 128);
  bn_finalize<<<1, 128, 0, stream>>>(stats, g3, bt3, scale, shift, (float)NN);

  // ---- edge update: new_e = (x3[row]+x3[col]) @ Wu + bu + e3 ----
  //      (pair-gather fused into A-stage; output overwrites eh buffer)
  gemm_wmma_f32<<<gE, gemmThreads, 0, stream>>>(bufB, Wu, bu, nullptr, nullptr,
                                                rowi, coli, e3,
                                                eh, EE, HH);

  // ---- x_out = bn3(relu(x3)) @ Wn + bn  (bn3 fused; uses scale/shift now) ----
  gemm_wmma_f32<<<gN, gemmThreads, 0, stream>>>(bufB, Wn, bn_b, scale, shift,
                                                nullptr, nullptr, nullptr,
                                                x_out, NN, HH);

  // ---- bn_edge3 stats on relu(new_e) ----
  hipMemsetAsync(stats, 0, 256 * sizeof(float), stream);
  bn_stats<<<(EE + 511) / 512, 256, 0, stream>>>(eh, stats, EE, 512);
  bn_finalize<<<1, 128, 0, stream>>>(stats, ge, bte, scale, shift, (float)EE);

  // ---- e_out = bn_edge3(relu(new_e)) @ Weo + beo  (bn fused) ----
  gemm_wmma_f32<<<gE, gemmThreads, 0, stream>>>(eh, Weo, beo, scale, shift,
                                                nullptr, nullptr, nullptr,
                                                e_out, EE, HH);
}